// RawToRgb_49185965473914
// MI455X (gfx1250) — compile-verified
//
#include <hip/hip_runtime.h>
#include <cstdint>

// ---------------------------------------------------------------------------
// Bayer (BG, top-left red) demosaic: (N,1,2048,2048) f32 -> (N,3,2048,2048) f32
// Memory-bound (512 MB @ 23.3 TB/s ~ 22 us floor). Strategy: stage raw tile
// (+reflect-101 halo) into LDS once via CDNA5 async global->LDS DMA path,
// compute R/G/B 3x3 stencil from LDS, stream results out with NT b128 stores.
// ---------------------------------------------------------------------------

namespace {
constexpr int IH = 2048;
constexpr int IW = 2048;
constexpr int TH = 16;                 // output tile rows per block
constexpr int TW = 128;                // output tile cols per block
constexpr int HROWS = TH + 2;          // 18 rows incl. halo
constexpr int LSTRIDE = TW + 8;        // 136 floats = 544 B (16B multiple)
constexpr int NWAVES = 8;              // wave32 x 8 = 256 threads
constexpr int NTHREADS = NWAVES * 32;
}

typedef float vfloat4 __attribute__((ext_vector_type(4)));

// reflect-101: -1 -> 1, n -> n-2 (only +/-1 excursions occur here)
__device__ __forceinline__ int refl101(int i, int n) {
  return (i < 0) ? -i : ((i >= n) ? (2 * n - 2 - i) : i);
}

// ---- CDNA5 async global->LDS copy (ASYNCcnt-tracked) ----------------------
// Inline asm: VFLAT form "global_load_async_to_lds_bN vdst(LDS addr), vaddr, off".
// LDS byte offset = low 32 bits of the generic pointer (aperture bits are in
// the high dword per the ISA flat-addressing rules).
__device__ __forceinline__ void async_g2l_b128(const float* g, float* l) {
#if defined(__HIP_DEVICE_COMPILE__)
  asm volatile("global_load_async_to_lds_b128 %0, %1, off"
               :: "v"((unsigned)(uintptr_t)l), "v"(g)
               : "memory");
#endif
}

__device__ __forceinline__ void async_g2l_b32(const float* g, float* l) {
#if defined(__HIP_DEVICE_COMPILE__)
  asm volatile("global_load_async_to_lds_b32 %0, %1, off"
               :: "v"((unsigned)(uintptr_t)l), "v"(g)
               : "memory");
#endif
}

__device__ __forceinline__ void wait_async_zero() {
#if defined(__HIP_DEVICE_COMPILE__)
#if __has_builtin(__builtin_amdgcn_s_wait_asynccnt)
  __builtin_amdgcn_s_wait_asynccnt(0);
#else
  asm volatile("s_wait_asynccnt 0" ::: "memory");
#endif
#endif
}

__device__ __forceinline__ void nt_store4(float* p, float a, float b, float c,
                                          float d) {
  vfloat4 v = {a, b, c, d};
  __builtin_nontemporal_store(v, (vfloat4*)p);
}

__global__ __launch_bounds__(NTHREADS)
void demosaic_bg_kernel(const float* __restrict__ img, float* __restrict__ out) {
  __shared__ float tile[HROWS * LSTRIDE];

  const int x0 = blockIdx.x * TW;
  const int y0 = blockIdx.y * TH;        // even (TH=16)
  const int n = blockIdx.z;
  const int tid = threadIdx.x;
  const int wave = tid >> 5;
  const int lane = tid & 31;

  const float* img_n = img + (size_t)n * IH * IW;

  // ---- stage tile + halo into LDS via async DMA (reflect-101 indexing) ----
  for (int r = wave; r < HROWS; r += NWAVES) {
    const int gy = refl101(y0 - 1 + r, IH);
    const float* grow = img_n + (size_t)gy * IW;
    // 128-float body: 32 lanes x 16B, LDS dst 16B-aligned (col offset 4)
    async_g2l_b128(grow + x0 + lane * 4, &tile[r * LSTRIDE + 4 + lane * 4]);
    // halo columns (left at LDS col 3, right at LDS col TW+4)
    if (lane < 2) {
      const int gx = (lane == 0) ? refl101(x0 - 1, IW) : refl101(x0 + TW, IW);
      const int lc = (lane == 0) ? 3 : (TW + 4);
      async_g2l_b32(grow + gx, &tile[r * LSTRIDE + lc]);
    }
  }
  wait_async_zero();
  __syncthreads();

  const size_t plane = (size_t)IH * IW;
  float* out_n = out + (size_t)n * 3 * plane;

  // ---- compute: each thread does two 4-wide pixel groups ------------------
#pragma unroll
  for (int p = 0; p < (TH * TW) / (NTHREADS * 4); ++p) {
    const int pid = tid + p * NTHREADS;   // 0..511
    const int ly = pid >> 5;              // 0..15 (wave-uniform)
    const int lx0 = (pid & 31) << 2;      // 0..124 step 4
    const int py = ly & 1;                // global row parity (y0 even)

    // row pointers; index k=0..5 covers cols lx0-1 .. lx0+4 (halo offset +4)
    const float* rm = &tile[(ly + 0) * LSTRIDE + lx0 + 3];
    const float* rc = &tile[(ly + 1) * LSTRIDE + lx0 + 3];
    const float* rd = &tile[(ly + 2) * LSTRIDE + lx0 + 3];

    float u[6], c[6], d[6];
#pragma unroll
    for (int k = 0; k < 6; ++k) { u[k] = rm[k]; c[k] = rc[k]; d[k] = rd[k]; }

    float rv[4], gv[4], bv[4];
#pragma unroll
    for (int j = 0; j < 4; ++j) {         // x parity == j&1 (lx0 % 4 == 0)
      const float center = c[j + 1];
      const float vsum = u[j + 1] + d[j + 1];
      const float hsum = c[j] + c[j + 2];
      const float cross = 0.25f * (vsum + hsum);
      const float diag = 0.25f * (u[j] + u[j + 2] + d[j] + d[j + 2]);
      const float horiz = 0.5f * hsum;
      const float vert = 0.5f * vsum;
      if (py == 0) {
        if ((j & 1) == 0) { rv[j] = center; gv[j] = cross;  bv[j] = diag; }
        else              { rv[j] = horiz;  gv[j] = center; bv[j] = vert; }
      } else {
        if ((j & 1) == 0) { rv[j] = vert;   gv[j] = center; bv[j] = horiz; }
        else              { rv[j] = diag;   gv[j] = cross;  bv[j] = center; }
      }
    }

    float* o = out_n + (size_t)(y0 + ly) * IW + (x0 + lx0);
    nt_store4(o,             rv[0], rv[1], rv[2], rv[3]);   // R plane
    nt_store4(o + plane,     gv[0], gv[1], gv[2], gv[3]);   // G plane
    nt_store4(o + 2 * plane, bv[0], bv[1], bv[2], bv[3]);   // B plane
  }
}

extern "C" void kernel_launch(void* const* d_in, const int* in_sizes, int n_in,
                              void* d_out, int out_size, void* d_ws,
                              size_t ws_size, hipStream_t stream) {
  const float* img = (const float*)d_in[0];
  float* out = (float*)d_out;
  const int n_imgs = in_sizes[0] / (IH * IW);   // 8 for the reference shapes
  dim3 grid(IW / TW, IH / TH, n_imgs);
  demosaic_bg_kernel<<<grid, dim3(NTHREADS), 0, stream>>>(img, out);
}